// SparseVLMModel_59725815218738
// MI455X (gfx1250) — compile-verified
//
#include <hip/hip_runtime.h>
#include <hip/hip_bf16.h>

#define TOP_K   32
#define D_VIS   768
#define D_LLM   4096
#define B_SZ    64
#define S_TOK   197
#define S_PATCH 196
#define T_TXT   512
#define OUT_S   (TOP_K + T_TXT)   // 544 rows per batch in output
#define LN_EPS  1e-5f

typedef float v2f __attribute__((ext_vector_type(2)));
typedef float v8f __attribute__((ext_vector_type(8)));

// ---------------------------------------------------------------------------
// K1: fused LayerNorm + scorer dot product. One wave32 per (b,s) row.
// score = inv*(sum(x*g*w) - mean*sum(g*w)) + sum(beta*w) + scorer_b
// ---------------------------------------------------------------------------
__global__ void __launch_bounds__(256)
ln_score_kernel(const float* __restrict__ vis,
                const float* __restrict__ gamma,
                const float* __restrict__ beta,
                const float* __restrict__ sw,
                const float* __restrict__ sb,      // scalar (1 elem)
                float* __restrict__ scores) {
    int wave = (blockIdx.x * blockDim.x + threadIdx.x) >> 5;
    int lane = threadIdx.x & 31;
    if (wave >= B_SZ * S_PATCH) return;
    int b = wave / S_PATCH;
    int s = wave - b * S_PATCH;
    const float* row = vis + ((size_t)b * S_TOK + 1 + s) * D_VIS;

    float s1 = 0.f, s2 = 0.f, sxgw = 0.f, sgw = 0.f, sbw = 0.f;
    #pragma unroll 4
    for (int k = lane; k < D_VIS; k += 32) {
        float x  = row[k];
        float gw = gamma[k] * sw[k];
        s1   += x;
        s2   += x * x;
        sxgw += x * gw;
        sgw  += gw;
        sbw  += beta[k] * sw[k];
    }
    #pragma unroll
    for (int off = 16; off > 0; off >>= 1) {
        s1   += __shfl_xor(s1,   off);
        s2   += __shfl_xor(s2,   off);
        sxgw += __shfl_xor(sxgw, off);
        sgw  += __shfl_xor(sgw,  off);
        sbw  += __shfl_xor(sbw,  off);
    }
    if (lane == 0) {
        const float rn = 1.0f / (float)D_VIS;
        float mean = s1 * rn;
        float var  = s2 * rn - mean * mean;
        float inv  = rsqrtf(var + LN_EPS);
        scores[wave] = inv * (sxgw - mean * sgw) + sbw + sb[0];
    }
}

// ---------------------------------------------------------------------------
// K2: exact top-k via ranking (matches jax.lax.top_k ordering & tie-break).
// One block per batch, 196 scores in LDS, O(196^2) comparisons.
// ---------------------------------------------------------------------------
__global__ void __launch_bounds__(256)
topk_kernel(const float* __restrict__ scores, int* __restrict__ topk) {
    __shared__ float sm[S_PATCH];
    int b = blockIdx.x;
    int i = threadIdx.x;
    if (i < S_PATCH) sm[i] = scores[b * S_PATCH + i];
    __syncthreads();
    if (i < S_PATCH) {
        float si = sm[i];
        int rank = 0;
        for (int j = 0; j < S_PATCH; ++j) {
            float sj = sm[j];
            rank += (sj > si) || ((sj == si) && (j < i));
        }
        if (rank < TOP_K) topk[b * TOP_K + rank] = i;
    }
}

// ---------------------------------------------------------------------------
// K3: gathered projection GEMM with V_WMMA_F32_16X16X4_F32 (exact fp32).
// One wave per 16x16 output tile: grid (4096/16 ntiles, 2 mtiles, 64 batch).
// A fragment (16x4): lanes 0-15 hold K=0,1 of row M=lane; lanes 16-31 K=2,3.
// B fragment (4x16): lanes 0-15 hold N=lane at K=0(v0),1(v1); lanes 16-31 K=2,3.
// C/D (16x16 f32): VGPR r -> M = (lane<16 ? r : 8+r), N = lane&15.
// Writes rows [0,32) of each batch's 544-row output slab (bias fused).
// ---------------------------------------------------------------------------
__global__ void __launch_bounds__(32)
proj_wmma_kernel(const float* __restrict__ vis,
                 const int* __restrict__ topk,
                 const float* __restrict__ W,     // [D_VIS, D_LLM] row-major
                 const float* __restrict__ pb,    // [D_LLM]
                 float* __restrict__ out) {
    const int nt   = blockIdx.x;          // 0..255
    const int mt   = blockIdx.y;          // 0..1
    const int b    = blockIdx.z;          // 0..63
    const int lane = threadIdx.x;
    const int lm   = lane & 15;
    const int kh   = (lane >> 4) << 1;    // 0 or 2: K sub-offset for this half-wave

    // Gathered A row for this lane's M index (through top-k indices).
    const int srow = topk[b * TOP_K + mt * 16 + lm];
    const float* aptr = vis + ((size_t)b * S_TOK + 1 + srow) * D_VIS + kh;
    // B: row-major weights, coalesced across lanes.
    const float* bptr = W + (size_t)kh * D_LLM + nt * 16 + lm;

    v8f acc = {};
    #pragma unroll 4
    for (int k0 = 0; k0 < D_VIS; k0 += 4) {
        v2f a = *(const v2f*)(aptr + k0);            // 8B-aligned pair load
        v2f bb;
        bb.x = bptr[(size_t)k0 * D_LLM];
        bb.y = bptr[(size_t)(k0 + 1) * D_LLM];
        acc = __builtin_amdgcn_wmma_f32_16x16x4_f32(
            /*neg_a=*/false, a, /*neg_b=*/false, bb,
            /*c_mod=*/(short)0, acc, /*reuse_a=*/false, /*reuse_b=*/false);
    }

    const int col   = nt * 16 + lm;
    const float bv  = pb[col];
    const int mbase = mt * 16 + (lane >> 4) * 8;
    float* ocol = out + (size_t)b * OUT_S * D_LLM + col;
    #pragma unroll
    for (int r = 0; r < 8; ++r)
        ocol[(size_t)(mbase + r) * D_LLM] = acc[r] + bv;
}

// ---------------------------------------------------------------------------
// K4: embedding gather + concat. One block per (b,t) row; float4 streaming.
// ---------------------------------------------------------------------------
__global__ void __launch_bounds__(256)
embed_kernel(const int* __restrict__ ids,
             const float* __restrict__ table,
             float* __restrict__ out) {
    int row = blockIdx.x;                 // 0 .. B_SZ*T_TXT-1
    int b = row >> 9;                     // /512
    int t = row & (T_TXT - 1);
    int id = ids[(size_t)b * T_TXT + t];
    const float4* src = (const float4*)(table + (size_t)id * D_LLM);
    float4* dst = (float4*)(out + ((size_t)b * OUT_S + TOP_K + t) * D_LLM);
    #pragma unroll
    for (int i = threadIdx.x; i < D_LLM / 4; i += 256)
        dst[i] = src[i];
}

// ---------------------------------------------------------------------------
extern "C" void kernel_launch(void* const* d_in, const int* in_sizes, int n_in,
                              void* d_out, int out_size, void* d_ws, size_t ws_size,
                              hipStream_t stream) {
    const float* vis   = (const float*)d_in[0];   // (64,197,768)
    const int*   ids   = (const int*)  d_in[1];   // (64,512)
    const float* gamma = (const float*)d_in[2];   // (768,)
    const float* beta  = (const float*)d_in[3];   // (768,)
    const float* sw    = (const float*)d_in[4];   // (768,)
    const float* sb    = (const float*)d_in[5];   // scalar
    const float* pw    = (const float*)d_in[6];   // (768,4096)
    const float* pb    = (const float*)d_in[7];   // (4096,)
    const float* tab   = (const float*)d_in[8];   // (32000,4096)
    float* out = (float*)d_out;                   // (64,544,4096)

    float* scores = (float*)d_ws;                                   // 64*196 f32
    int*   topk   = (int*)((char*)d_ws + B_SZ * S_PATCH * sizeof(float)); // 64*32 i32

    const int rows = B_SZ * S_PATCH;                 // 12544 waves
    ln_score_kernel<<<(rows + 7) / 8, 256, 0, stream>>>(vis, gamma, beta, sw, sb, scores);
    topk_kernel<<<B_SZ, 256, 0, stream>>>(scores, topk);
    proj_wmma_kernel<<<dim3(D_LLM / 16, 2, B_SZ), 32, 0, stream>>>(vis, topk, pw, pb, out);
    embed_kernel<<<B_SZ * T_TXT, 256, 0, stream>>>(ids, tab, out);
}